// TimesBlock_58677843198548
// MI455X (gfx1250) — compile-verified
//
#include <hip/hip_runtime.h>

// ---------------------------------------------------------------------------
// CDNA5 / gfx1250: everything expressed as f32 GEMMs on v_wmma_f32_16x16x4_f32,
// with double-buffered global_load_async_to_lds_b32 staging (ASYNCcnt pipelined).
// ---------------------------------------------------------------------------

typedef float v2f __attribute__((ext_vector_type(2)));
typedef float v8f __attribute__((ext_vector_type(8)));

#define Bdim   128
#define Ndim   128
#define Tdim   720
#define PER    24
#define PIN    30
#define POUT   30
#define EMB    64
#define FREQ   33
#define DOM    72
#define CH     (Bdim*Ndim)        // 16384 channels

#define K2PI   6.283185307179586

// GEMM tiling: 128x64 C-tile, K-step 16, 256 threads = 8 wave32 waves.
// Wave w: rows (w&3)*32..+31, cols (w>>2)*32..+31  -> 4 accumulators (2x2 tiles).
#define TM 128
#define TN 64
#define TK 16
#define APAD 17     // LDS A row stride (17 coprime 32 banks)
#define BPAD 65
#define SLACK 256   // float slack after each ws region (covers K-pad overreads)

// ---------------------------------------------------------------------------
// GEMM: C[M,N] = A[M,K(lda)] @ Bw[Kpad,N] (+bias[N]).
//  - M must be a multiple of 128 (true for all call sites).
//  - Kpad multiple of 16; Bw rows [lda..Kpad) must be zero (builders do this);
//    A overreads past row end multiply those zero rows -> no K guards needed.
//  - OOB N columns compute garbage but are never stored.
// Fragment layouts per CDNA5 ISA 7.12.2 (16x4 A / 4x16 B / 16x16 C).
// ---------------------------------------------------------------------------
__global__ __launch_bounds__(256)
void gemm_f32_wmma(const float* __restrict__ A, int lda,
                   const float* __restrict__ Bw,
                   const float* __restrict__ bias, float* __restrict__ C,
                   int M, int N, int Kpad)
{
    __shared__ float As[2][TM][APAD];
    __shared__ float Bs[2][TK][BPAD];

    const int tid  = threadIdx.x;
    const int lane = tid & 31;
    const int wave = tid >> 5;
    const int wr   = wave & 3;       // 32-row quadrant
    const int wc   = wave >> 2;      // 32-col quadrant
    const int half = lane >> 4;      // K-pair select within fragment
    const int lid  = lane & 15;

    const int m0 = blockIdx.y * TM;
    const int n0 = blockIdx.x * TN;

    const unsigned asBase = (unsigned)(size_t)(&As[0][0][0]);
    const unsigned bsBase = (unsigned)(size_t)(&Bs[0][0][0]);

    // per-thread async-load slots: A 128x16 = 8 issues, B 16x64 = 4 issues
    const int ka = tid & 15, mA = tid >> 4;   // A: (mA + 16j, ka)
    const int nb = tid & 63, kB = tid >> 6;   // B: (kB + 4j, nb)

    const int nk = Kpad >> 4;

    auto issue = [&](int kt, int buf) {
        const int k0 = kt * TK;
        const float* ga = A + (size_t)(m0 + mA) * lda + (k0 + ka);
        unsigned la = asBase + (unsigned)buf * (TM * APAD * 4)
                             + (unsigned)(mA * APAD + ka) * 4;
#pragma unroll
        for (int j = 0; j < 8; ++j) {
            asm volatile("global_load_async_to_lds_b32 %0, %1, off"
                         :: "v"(la), "v"(ga) : "memory");
            ga += (size_t)16 * lda;
            la += 16 * APAD * 4;
        }
        const float* gb = Bw + (size_t)(k0 + kB) * N + (n0 + nb);
        unsigned lb = bsBase + (unsigned)buf * (TK * BPAD * 4)
                             + (unsigned)(kB * BPAD + nb) * 4;
#pragma unroll
        for (int j = 0; j < 4; ++j) {
            asm volatile("global_load_async_to_lds_b32 %0, %1, off"
                         :: "v"(lb), "v"(gb) : "memory");
            gb += (size_t)4 * N;
            lb += 4 * BPAD * 4;
        }
    };

    v8f a00 = {}, a01 = {}, a10 = {}, a11 = {};

    issue(0, 0);
    for (int kt = 0; kt < nk; ++kt) {
        const int cur = kt & 1;
        if (kt + 1 < nk) {
            issue(kt + 1, cur ^ 1);
            asm volatile("s_wait_asynccnt 12" ::: "memory");  // tile kt landed
        } else {
            asm volatile("s_wait_asynccnt 0" ::: "memory");
        }
        __syncthreads();

        const int r0 = wr * 32 + lid, r1 = r0 + 16;
        const int c0 = wc * 32 + lid, c1 = c0 + 16;
#pragma unroll
        for (int kk = 0; kk < TK; kk += 4) {
            const int kb = kk + 2 * half;
            v2f fa0, fa1, fb0, fb1;
            fa0.x = As[cur][r0][kb];  fa0.y = As[cur][r0][kb + 1];
            fa1.x = As[cur][r1][kb];  fa1.y = As[cur][r1][kb + 1];
            fb0.x = Bs[cur][kb][c0];  fb0.y = Bs[cur][kb + 1][c0];
            fb1.x = Bs[cur][kb][c1];  fb1.y = Bs[cur][kb + 1][c1];
            a00 = __builtin_amdgcn_wmma_f32_16x16x4_f32(false, fa0, false, fb0, (short)0, a00, false, false);
            a01 = __builtin_amdgcn_wmma_f32_16x16x4_f32(false, fa0, false, fb1, (short)0, a01, false, false);
            a10 = __builtin_amdgcn_wmma_f32_16x16x4_f32(false, fa1, false, fb0, (short)0, a10, false, false);
            a11 = __builtin_amdgcn_wmma_f32_16x16x4_f32(false, fa1, false, fb1, (short)0, a11, false, false);
        }
        __syncthreads();
    }

    const int rowb = m0 + wr * 32;
    const int colb = n0 + wc * 32;
#pragma unroll
    for (int v = 0; v < 8; ++v) {
        int rr0 = rowb + v + 8 * half;
        int rr1 = rr0 + 16;
        int cc0 = colb + lid;
        int cc1 = cc0 + 16;
        if (cc0 < N) {
            float bz = bias ? bias[cc0] : 0.0f;
            C[(size_t)rr0 * N + cc0] = a00[v] + bz;
            C[(size_t)rr1 * N + cc0] = a10[v] + bz;
        }
        if (cc1 < N) {
            float bz = bias ? bias[cc1] : 0.0f;
            C[(size_t)rr0 * N + cc1] = a01[v] + bz;
            C[(size_t)rr1 * N + cc1] = a11[v] + bz;
        }
    }
}

// ---------------------------------------------------------------------------
// Batched tiled transpose: in[b][R][C] -> out[b][C][R].  block (32,8).
// ---------------------------------------------------------------------------
__global__ void transpose_batched(const float* __restrict__ in,
                                  float* __restrict__ out, int R, int C)
{
    __shared__ float tile[32][33];
    const int b  = blockIdx.z;
    const int r0 = blockIdx.y * 32;
    const int c0 = blockIdx.x * 32;
    const float* ib = in + (size_t)b * R * C;
    float* ob = out + (size_t)b * R * C;
    for (int i = threadIdx.y; i < 32; i += 8) {
        int r = r0 + i, c = c0 + threadIdx.x;
        tile[i][threadIdx.x] = (r < R && c < C) ? ib[(size_t)r * C + c] : 0.0f;
    }
    __syncthreads();
    for (int i = threadIdx.y; i < 32; i += 8) {
        int c = c0 + i, r = r0 + threadIdx.x;
        if (c < C && r < R) ob[(size_t)c * R + r] = tile[threadIdx.x][i];
    }
}

// [bn, p, Re33|Im33] (+W_pos into Re) -> [bn, f, Re30|Im30]
__global__ void swap_pf_to_fp(const float* __restrict__ in,
                              const float* __restrict__ wpos,
                              float* __restrict__ out, int total)
{
    int i = blockIdx.x * 256 + threadIdx.x;
    if (i >= total) return;
    int p  = i % PIN;
    int j  = i / PIN;
    int f  = j % FREQ;
    int bn = j / FREQ;
    const float* src = in + (size_t)(bn * PIN + p) * (2 * FREQ);
    float re = src[f];
    float im = src[FREQ + f];
    if (wpos) re += wpos[(size_t)(bn * PIN + p) * FREQ + f];
    float* dst = out + (size_t)(bn * FREQ + f) * (2 * PIN);
    dst[p]       = re;
    dst[PIN + p] = im;
}

// [bn, f, Re30|Im30] -> [bn, p, Re33|Im33]
__global__ void swap_fp_to_pf(const float* __restrict__ in,
                              float* __restrict__ out, int total)
{
    int i = blockIdx.x * 256 + threadIdx.x;
    if (i >= total) return;
    int f  = i % FREQ;
    int j  = i / FREQ;
    int p  = j % PIN;
    int bn = j / PIN;
    const float* src = in + (size_t)(bn * FREQ + f) * (2 * PIN);
    float* dst = out + (size_t)(bn * PIN + p) * (2 * FREQ);
    dst[f]        = src[p];
    dst[FREQ + f] = src[PIN + p];
}

// ---------------------------------------------------------------------------
// Weight builders (tiny, rebuilt every call; exact-integer phase for accuracy).
// All GEMM B-operands are zero-padded in K up to Kpad (multiple of 16).
// ---------------------------------------------------------------------------
__global__ void build_basis720(float* __restrict__ bw)     // [720 x 144]
{
    int i = blockIdx.x * 256 + threadIdx.x;
    if (i >= Tdim * 2 * DOM) return;
    int k2 = i % (2 * DOM);
    int t  = i / (2 * DOM);
    int k  = (k2 < DOM) ? k2 : (k2 - DOM);
    double a = K2PI * ((k * t) % Tdim) / Tdim;
    bw[i] = (k2 < DOM) ? (float)cos(a) : (float)(-sin(a));
}

__global__ void build_basisInv720(float* __restrict__ bw)  // [144 x 720]
{
    int i = blockIdx.x * 256 + threadIdx.x;
    if (i >= 2 * DOM * Tdim) return;
    int t  = i % Tdim;
    int kk = i / Tdim;
    int k  = (kk < DOM) ? kk : (kk - DOM);
    double a = K2PI * ((k * t) % Tdim) / Tdim;
    double v;
    if (kk < DOM) v = (k == 0 ? 1.0 : 2.0 * cos(a)) / Tdim;
    else          v = (k == 0 ? 0.0 : -2.0 * sin(a) / Tdim);
    bw[i] = (float)v;
}

// complex linear as real block matrix [Kpad x 2D]; +I folds residual.
__global__ void build_wc_complex(const float* __restrict__ Wr,
                                 const float* __restrict__ Wi,
                                 const float* __restrict__ br,
                                 const float* __restrict__ bi,
                                 float* __restrict__ wc, float* __restrict__ bias,
                                 int D, int addI, int Kpad)
{
    int i = blockIdx.x * 256 + threadIdx.x;
    int W = 2 * D;
    int tot = Kpad * W;
    if (i < tot) {
        int nn = i % W;
        int kk = i / W;
        float v = 0.0f;
        if (kk < W) {
            int ii = (kk < D) ? kk : kk - D;  int reIn  = (kk < D);
            int oo = (nn < D) ? nn : nn - D;  int reOut = (nn < D);
            float wr = Wr[oo * D + ii], wi = Wi[oo * D + ii];
            if (reIn && reOut)        v = wr + ((addI && ii == oo) ? 1.0f : 0.0f);
            else if (!reIn && reOut)  v = -wi;
            else if (reIn && !reOut)  v = wi;
            else                      v = wr + ((addI && ii == oo) ? 1.0f : 0.0f);
        }
        wc[i] = v;
    } else if (i < tot + W) {
        int j = i - tot;
        bias[j] = (j < D) ? br[j] : bi[j - D];
    }
}

// embedding (24->64) composed with 64-pt rfft: [32 x 66] (rows 24..31 zero) + bias
__global__ void build_embfft(const float* __restrict__ WPW,
                             const float* __restrict__ WPb,
                             float* __restrict__ wemb, float* __restrict__ bemb)
{
    int i = blockIdx.x * 256 + threadIdx.x;
    if (i >= 33 * 66) return;
    int c = i % 66;
    int r = i / 66;              // r==32 -> bias row
    int f = (c < FREQ) ? c : c - FREQ;
    int isRe = (c < FREQ);
    if (r >= 24 && r < 32) { wemb[r * 66 + c] = 0.0f; return; }
    double s = 0.0;
    for (int o = 0; o < EMB; ++o) {
        double a = K2PI * ((f * o) & (EMB - 1)) / EMB;
        double w = (r < 24) ? (double)WPW[o * PER + r] : (double)WPb[o];
        s += w * (isRe ? cos(a) : -sin(a));
    }
    if (r < 24) wemb[r * 66 + c] = (float)s;
    else        bemb[c] = (float)s;
}

// 64-pt irfft basis [80 x 64] (rows 66..79 zero); bin 32 = Nyquist
__global__ void build_basis64(float* __restrict__ bw)
{
    int i = blockIdx.x * 256 + threadIdx.x;
    if (i >= 80 * EMB) return;
    int e  = i % EMB;
    int rr = i / EMB;
    if (rr >= 66) { bw[i] = 0.0f; return; }
    int f  = (rr < FREQ) ? rr : rr - FREQ;
    double a = K2PI * ((f * e) & (EMB - 1)) / EMB;
    double v;
    if (rr < FREQ) v = (f == 0 ? 1.0 : (f == 32 ? cos(a) : 2.0 * cos(a))) / EMB;
    else           v = (f == 0 || f == 32) ? 0.0 : -2.0 * sin(a) / EMB;
    bw[i] = (float)v;
}

// out[b,t,n] = 0.7*pred[bn,t] + 0.3*low[bn,t]   (tiled transpose store)
__global__ void combine_out(const float* __restrict__ pred,
                            const float* __restrict__ low,
                            float* __restrict__ out)
{
    __shared__ float tile[32][33];
    const int b  = blockIdx.z;
    const int n0 = blockIdx.y * 32;
    const int t0 = blockIdx.x * 32;
    for (int i = threadIdx.y; i < 32; i += 8) {
        int n = n0 + i, t = t0 + threadIdx.x;
        float v = 0.0f;
        if (t < Tdim) {
            size_t idx = (size_t)(b * Ndim + n) * Tdim + t;
            v = 0.7f * pred[idx] + 0.3f * low[idx];
        }
        tile[i][threadIdx.x] = v;
    }
    __syncthreads();
    for (int i = threadIdx.y; i < 32; i += 8) {
        int t = t0 + i, n = n0 + threadIdx.x;
        if (t < Tdim) out[(size_t)(b * Tdim + t) * Ndim + n] = tile[threadIdx.x][i];
    }
}

// ---------------------------------------------------------------------------
// Host orchestration
// ---------------------------------------------------------------------------
static inline dim3 gemm_grid(int M, int N) {
    return dim3((N + TN - 1) / TN, M / TM);
}

extern "C" void kernel_launch(void* const* d_in, const int* in_sizes, int n_in,
                              void* d_out, int out_size, void* d_ws, size_t ws_size,
                              hipStream_t stream)
{
    const float* x       = (const float*)d_in[0];
    const float* freq_Wr = (const float*)d_in[1];
    const float* freq_Wi = (const float*)d_in[2];
    const float* freq_br = (const float*)d_in[3];
    const float* freq_bi = (const float*)d_in[4];
    const float* WP_W    = (const float*)d_in[5];
    const float* WP_b    = (const float*)d_in[6];
    const float* W_pos   = (const float*)d_in[7];
    const float* mix1_Wr = (const float*)d_in[8];
    const float* mix1_Wi = (const float*)d_in[9];
    const float* mix1_br = (const float*)d_in[10];
    const float* mix1_bi = (const float*)d_in[11];
    const float* fmix_Wr = (const float*)d_in[12];
    const float* fmix_Wi = (const float*)d_in[13];
    const float* fmix_br = (const float*)d_in[14];
    const float* fmix_bi = (const float*)d_in[15];
    const float* proj_Wr = (const float*)d_in[16];
    const float* proj_Wi = (const float*)d_in[17];
    const float* proj_br = (const float*)d_in[18];
    const float* proj_bi = (const float*)d_in[19];
    const float* head_W  = (const float*)d_in[20];
    const float* head_b  = (const float*)d_in[21];
    float* out = (float*)d_out;

    // workspace carve-up (floats); SLACK after each region covers K-pad overreads
    float* w = (float*)d_ws;
    float* xT    = w; w += (size_t)CH * Tdim + SLACK;          // [16384,720]
    float* lowxy = w; w += (size_t)CH * Tdim + SLACK;          // [16384,720]
    float* Lri   = w; w += (size_t)CH * 2 * DOM + SLACK;       // [16384,144]
    float* L2    = w; w += (size_t)CH * 2 * DOM + SLACK;       // [16384,144]
    float* buf0  = w; w += (size_t)CH * PIN * 2 * FREQ + SLACK;  // 32.44M floats
    float* buf1  = w; w += (size_t)CH * PIN * 2 * FREQ + SLACK;  // 32.44M floats
    float* bas720 = w; w += Tdim * 2 * DOM + SLACK;            // [720,144]
    float* wcFreq = w; w += 2 * DOM * 2 * DOM + SLACK;         // [144,144]
    float* bFreq  = w; w += 2 * DOM + SLACK;
    float* basInv = w; w += 2 * DOM * Tdim + SLACK;            // [144,720]
    float* wEmb   = w; w += 32 * 66 + SLACK;                   // [32,66] K-padded
    float* bEmb   = w; w += 66 + SLACK;
    float* wcM1   = w; w += 64 * 60 + SLACK;  float* bM1 = w; w += 60 + SLACK;
    float* wcFm   = w; w += 80 * 66 + SLACK;  float* bFm = w; w += 66 + SLACK;
    float* wcPj   = w; w += 64 * 60 + SLACK;  float* bPj = w; w += 60 + SLACK;
    float* bas64  = w; w += 80 * EMB + SLACK;                  // [80,64] K-padded
    float* headWt = w; w += (size_t)(POUT * EMB) * Tdim + SLACK; // [1920,720]

    const dim3 blk256(256);
    const dim3 tblk(32, 8);

    // ---- weight / basis preparation (tiny) ----
    transpose_batched<<<dim3(4, 23, Bdim), tblk, 0, stream>>>(x, xT, Tdim, Ndim);
    transpose_batched<<<dim3(60, 23, 1), tblk, 0, stream>>>(head_W, headWt, Tdim, POUT * EMB);
    build_basis720<<<(Tdim * 2 * DOM + 255) / 256, blk256, 0, stream>>>(bas720);
    build_basisInv720<<<(2 * DOM * Tdim + 255) / 256, blk256, 0, stream>>>(basInv);
    build_wc_complex<<<(144 * 144 + 144 + 255) / 256, blk256, 0, stream>>>(
        freq_Wr, freq_Wi, freq_br, freq_bi, wcFreq, bFreq, DOM, 0, 144);
    build_embfft<<<(33 * 66 + 255) / 256, blk256, 0, stream>>>(WP_W, WP_b, wEmb, bEmb);
    build_wc_complex<<<(64 * 60 + 60 + 255) / 256, blk256, 0, stream>>>(
        mix1_Wr, mix1_Wi, mix1_br, mix1_bi, wcM1, bM1, PIN, 1, 64);
    build_wc_complex<<<(80 * 66 + 66 + 255) / 256, blk256, 0, stream>>>(
        fmix_Wr, fmix_Wi, fmix_br, fmix_bi, wcFm, bFm, FREQ, 1, 80);
    build_wc_complex<<<(64 * 60 + 60 + 255) / 256, blk256, 0, stream>>>(
        proj_Wr, proj_Wi, proj_br, proj_bi, wcPj, bPj, PIN, 0, 64);
    build_basis64<<<(80 * EMB + 255) / 256, blk256, 0, stream>>>(bas64);

    // ---- low-frequency branch: DFT -> complex linear -> iDFT ----
    gemm_f32_wmma<<<gemm_grid(CH, 2 * DOM), blk256, 0, stream>>>(
        xT, Tdim, bas720, nullptr, Lri, CH, 2 * DOM, Tdim);
    gemm_f32_wmma<<<gemm_grid(CH, 2 * DOM), blk256, 0, stream>>>(
        Lri, 2 * DOM, wcFreq, bFreq, L2, CH, 2 * DOM, 2 * DOM);
    gemm_f32_wmma<<<gemm_grid(CH, Tdim), blk256, 0, stream>>>(
        L2, 2 * DOM, basInv, nullptr, lowxy, CH, Tdim, 2 * DOM);

    // ---- period branch ----
    const int Mp = CH * PIN;      // 491520 rows [bn,p]
    const int Mf = CH * FREQ;     // 540672 rows [bn,f]
    const int totEl = CH * FREQ * PIN;

    // embedding + 64-pt rfft fused (A = xT reinterpreted as [491520,24])
    gemm_f32_wmma<<<gemm_grid(Mp, 66), blk256, 0, stream>>>(
        xT, PER, wEmb, bEmb, buf0, Mp, 66, 32);
    swap_pf_to_fp<<<(totEl + 255) / 256, blk256, 0, stream>>>(buf0, W_pos, buf1, totEl);
    // mix1 over p (+residual folded into weight)
    gemm_f32_wmma<<<gemm_grid(Mf, 60), blk256, 0, stream>>>(
        buf1, 60, wcM1, bM1, buf0, Mf, 60, 64);
    swap_fp_to_pf<<<(totEl + 255) / 256, blk256, 0, stream>>>(buf0, buf1, totEl);
    // fmix over f (+residual)
    gemm_f32_wmma<<<gemm_grid(Mp, 66), blk256, 0, stream>>>(
        buf1, 66, wcFm, bFm, buf0, Mp, 66, 80);
    swap_pf_to_fp<<<(totEl + 255) / 256, blk256, 0, stream>>>(buf0, nullptr, buf1, totEl);
    // proj over p
    gemm_f32_wmma<<<gemm_grid(Mf, 60), blk256, 0, stream>>>(
        buf1, 60, wcPj, bPj, buf0, Mf, 60, 64);
    swap_fp_to_pf<<<(totEl + 255) / 256, blk256, 0, stream>>>(buf0, buf1, totEl);
    // 64-pt irfft -> pp [16384, 1920]
    gemm_f32_wmma<<<gemm_grid(Mp, EMB), blk256, 0, stream>>>(
        buf1, 66, bas64, nullptr, buf0, Mp, EMB, 80);
    // head GEMM: [16384,1920] @ [1920,720] + head_b  (dominant, 45 GFLOP)
    gemm_f32_wmma<<<gemm_grid(CH, Tdim), blk256, 0, stream>>>(
        buf0, POUT * EMB, headWt, head_b, buf1, CH, Tdim, POUT * EMB);

    // ---- blend + transpose to [B, T, N] ----
    combine_out<<<dim3(23, 4, Bdim), tblk, 0, stream>>>(buf1, lowxy, out);
}